// BPSymmFunc_39539468927509
// MI455X (gfx1250) — compile-verified
//
#include <hip/hip_runtime.h>
#include <stdint.h>

#define NSF 8
#define RC_F 6.0f
#define LOG2E_F 1.4426950408889634f
#define PI_F 3.14159265358979323846f
#define TRIP_TILE 4096

typedef unsigned int v4u __attribute__((ext_vector_type(4)));
typedef int v8i __attribute__((ext_vector_type(8)));
typedef int v4i __attribute__((ext_vector_type(4)));

__device__ __forceinline__ float cutoff_fc(float d) {
  return (d < RC_F) ? 0.5f * (__cosf(d * (PI_F / RC_F)) + 1.0f) : 0.0f;
}

// ---------------------------------------------------------------------------
// Pair stage: dist/fc/G2 fingerprints + per-pair workspace for the G4 stage.
//   pA[p] = (ux, uy, uz, fc)   (unit direction + cutoff)
//   pM[p] = d^2 * log2(e)      (pre-scaled for exp2-based G4 exponent)
// ---------------------------------------------------------------------------
__global__ void __launch_bounds__(256)
bp_pair_kernel(const float* __restrict__ diff,
               const int* __restrict__ ind2,
               const float* __restrict__ Rs,
               const float* __restrict__ eta2,
               float* __restrict__ out,
               float4* __restrict__ pA,
               float* __restrict__ pM,
               int np, int storeWs)
{
  int p = blockIdx.x * blockDim.x + threadIdx.x;
  if (p >= np) return;
  float dx = diff[3 * p + 0], dy = diff[3 * p + 1], dz = diff[3 * p + 2];
  float d2 = fmaf(dx, dx, fmaf(dy, dy, dz * dz));
  float d = sqrtf(d2);
  float invd = 1.0f / d;
  float fc = cutoff_fc(d);
  if (storeWs) {
    pA[p] = make_float4(dx * invd, dy * invd, dz * invd, fc);
    pM[p] = d2 * LOG2E_F;
  }
  int ai = ind2[2 * p];
  float* dst = out + (size_t)ai * (2 * NSF);
#pragma unroll
  for (int s = 0; s < NSF; ++s) {
    float t = d - Rs[s];
    float v = __expf(-eta2[s] * t * t) * fc;
    atomicAdd(dst + s, v);
  }
}

// ---------------------------------------------------------------------------
// TDM: DMA a 1-D tile of int32 (the ind_3 stream) into LDS.
// D# per CDNA5 ISA ch.8:
//   group0: count=1 | lds_addr | global_addr(tile start) | type=2
//   group1: data_size=4B, tensor_dim0=tile_dim0=nelem (no OOB possible),
//           tensor_dim1=tile_dim1=1, stride0=nelem
// 6-arg builtin form (clang-23 / therock-10.0 headers).
// ---------------------------------------------------------------------------
__device__ __forceinline__ void tdm_load_i32_tile(uint32_t lds_byte_off,
                                                  const int* __restrict__ gsrc,
                                                  uint32_t nelem)
{
  uint64_t ga = (uint64_t)(uintptr_t)gsrc;
  v4u g0;
  g0[0] = 1u;                                                  // count=1, user D#
  g0[1] = lds_byte_off;                                        // lds_addr
  g0[2] = (uint32_t)(ga & 0xFFFFFFFFull);                      // global_addr lo
  g0[3] = (uint32_t)((ga >> 32) & 0x01FFFFFFull) | (2u << 30); // addr hi | type=2
  v8i g1;
  g1[0] = (int)(2u << 16);                                     // data_size = 4B
  g1[1] = (int)((nelem & 0xFFFFu) << 16);                      // tensor_dim0[15:0]
  g1[2] = (int)((nelem >> 16) | (1u << 16));                   // dim0 hi | tensor_dim1=1
  g1[3] = (int)((nelem & 0xFFFFu) << 16);                      // tile_dim0 = nelem
  g1[4] = (int)1u;                                             // tile_dim1=1, tile_dim2=0
  g1[5] = (int)nelem;                                          // tensor_dim0_stride lo
  g1[6] = 0;
  g1[7] = 0;
  v4i gz4 = {0, 0, 0, 0};
  v8i gz8 = {0, 0, 0, 0, 0, 0, 0, 0};
  __builtin_amdgcn_tensor_load_to_lds(g0, g1, gz4, gz4, gz8, 0);
}

// ---------------------------------------------------------------------------
// Triplet stage, TDM double-buffered. Each block pipelines tiles of
// TRIP_TILE triplets: wave 0 issues the next tile's tensor_load_to_lds,
// waits s_wait_tensorcnt<=1 (current tile landed), barrier, all 8 waves
// consume the LDS tile while the TDM streams the next one.
// ---------------------------------------------------------------------------
__global__ void __launch_bounds__(256)
bp_trip_tdm_kernel(const float4* __restrict__ pA,
                   const float* __restrict__ pM,
                   const int* __restrict__ ind2,
                   const int* __restrict__ ind3,
                   const float* __restrict__ lambd,
                   const float* __restrict__ zeta,
                   const float* __restrict__ eta4,
                   float* __restrict__ out,
                   int nt)
{
  __shared__ int buf[2][2 * TRIP_TILE]; // 64 KB (LDS is 320 KB/WGP)

  float lam[NSF], zet[NSF], et4[NSF], cc[NSF];
#pragma unroll
  for (int s = 0; s < NSF; ++s) {
    lam[s] = lambd[s];
    zet[s] = zeta[s];
    et4[s] = eta4[s];
    cc[s] = exp2f(1.0f - zet[s]); // 2^(1-zeta)
  }

  const bool lead = (threadIdx.x < 32); // wave 0 only (wave32) issues TDM
  int tilesTotal = (nt + TRIP_TILE - 1) / TRIP_TILE;
  int tile0 = (int)blockIdx.x;
  if (tile0 >= tilesTotal) return; // uniform across block

  int cur = 0;
  if (lead) {
    int base = tile0 * TRIP_TILE;
    uint32_t ne = (uint32_t)(2 * min(TRIP_TILE, nt - base));
    tdm_load_i32_tile((uint32_t)(uintptr_t)&buf[0][0], ind3 + 2 * (size_t)base, ne);
  }

  for (int tile = tile0; tile < tilesTotal; tile += (int)gridDim.x) {
    int nextTile = tile + (int)gridDim.x;
    if (lead) {
      if (nextTile < tilesTotal) {
        int nb = nextTile * TRIP_TILE;
        uint32_t ne = (uint32_t)(2 * min(TRIP_TILE, nt - nb));
        tdm_load_i32_tile((uint32_t)(uintptr_t)&buf[cur ^ 1][0],
                          ind3 + 2 * (size_t)nb, ne);
        __builtin_amdgcn_s_wait_tensorcnt((short)1); // current tile done
      } else {
        __builtin_amdgcn_s_wait_tensorcnt((short)0);
      }
    }
    __syncthreads();

    int base = tile * TRIP_TILE;
    int cnt = min(TRIP_TILE, nt - base);
    const int* __restrict__ tb = &buf[cur][0];
    for (int t = (int)threadIdx.x; t < cnt; t += 256) {
      int ij = tb[2 * t];
      int ik = tb[2 * t + 1];
      float4 a = pA[ij]; // L2-resident gathers
      float4 b = pA[ik];
      float cosT = fmaf(a.x, b.x, fmaf(a.y, b.y, a.z * b.z));
      float fcfc = a.w * b.w;
      float mm = pM[ij] + pM[ik]; // (Rij^2+Rik^2)*log2(e)
      int atom = ind2[2 * ij];
      float* dst = out + (size_t)atom * (2 * NSF) + NSF;
#pragma unroll
      for (int s = 0; s < NSF; ++s) {
        float bs = fmaxf(fmaf(lam[s], cosT, 1.0f), 0.0f);
        // base^zeta * exp(-eta*S) = exp2(zeta*log2(base) - eta*S*log2e)
        float e = fmaf(zet[s], __log2f(bs), -et4[s] * mm);
        float v = cc[s] * exp2f(e) * fcfc;
        atomicAdd(dst + s, v);
      }
    }
    __syncthreads();
    cur ^= 1;
  }
}

// ---------------------------------------------------------------------------
// Fallback triplet kernel (workspace too small): recompute from diff.
// ---------------------------------------------------------------------------
__global__ void __launch_bounds__(256)
bp_trip_direct_kernel(const float* __restrict__ diff,
                      const int* __restrict__ ind2,
                      const int* __restrict__ ind3,
                      const float* __restrict__ lambd,
                      const float* __restrict__ zeta,
                      const float* __restrict__ eta4,
                      float* __restrict__ out,
                      int nt)
{
  int t = blockIdx.x * blockDim.x + threadIdx.x;
  if (t >= nt) return;
  int ij = ind3[2 * t], ik = ind3[2 * t + 1];
  float ax = diff[3 * ij], ay = diff[3 * ij + 1], az = diff[3 * ij + 2];
  float bx = diff[3 * ik], by = diff[3 * ik + 1], bz = diff[3 * ik + 2];
  float d2i = fmaf(ax, ax, fmaf(ay, ay, az * az));
  float d2k = fmaf(bx, bx, fmaf(by, by, bz * bz));
  float di = sqrtf(d2i), dk = sqrtf(d2k);
  float cosT = fmaf(ax, bx, fmaf(ay, by, az * bz)) / (di * dk);
  float fcfc = cutoff_fc(di) * cutoff_fc(dk);
  float mm = (d2i + d2k) * LOG2E_F;
  int atom = ind2[2 * ij];
  float* dst = out + (size_t)atom * (2 * NSF) + NSF;
#pragma unroll
  for (int s = 0; s < NSF; ++s) {
    float bs = fmaxf(fmaf(lambd[s], cosT, 1.0f), 0.0f);
    float e = fmaf(zeta[s], __log2f(bs), -eta4[s] * mm);
    float v = exp2f(1.0f - zeta[s]) * exp2f(e) * fcfc;
    atomicAdd(dst + s, v);
  }
}

extern "C" void kernel_launch(void* const* d_in, const int* in_sizes, int n_in,
                              void* d_out, int out_size, void* d_ws, size_t ws_size,
                              hipStream_t stream)
{
  (void)n_in;
  const float* diff  = (const float*)d_in[0];
  // d_in[1] (elems) only provides n_atoms in the reference; values unused.
  const int*   ind2  = (const int*)d_in[2];
  const int*   ind3  = (const int*)d_in[3];
  const float* Rs    = (const float*)d_in[4];
  const float* eta2  = (const float*)d_in[5];
  const float* lambd = (const float*)d_in[6];
  const float* zeta  = (const float*)d_in[7];
  const float* eta4  = (const float*)d_in[8];
  int np = in_sizes[0] / 3;
  int nt = in_sizes[3] / 2;
  float* out = (float*)d_out;

  // Output is accumulated with atomics; zero it first (graph-capturable).
  (void)hipMemsetAsync(d_out, 0, (size_t)out_size * sizeof(float), stream);

  size_t needWs = (size_t)np * 16 + (size_t)np * 4; // pA + pM
  bool useWs = (d_ws != nullptr) && (ws_size >= needWs);
  float4* pA = (float4*)d_ws;
  float*  pM = (float*)((char*)d_ws + (size_t)np * 16);

  int pb = (np + 255) / 256;
  bp_pair_kernel<<<pb, 256, 0, stream>>>(diff, ind2, Rs, eta2, out, pA, pM, np,
                                         useWs ? 1 : 0);

  if (useWs) {
    int tiles = (nt + TRIP_TILE - 1) / TRIP_TILE;
    int nblk = tiles < 1024 ? tiles : 1024;
    bp_trip_tdm_kernel<<<nblk, 256, 0, stream>>>(pA, pM, ind2, ind3, lambd, zeta,
                                                 eta4, out, nt);
  } else {
    int tb = (nt + 255) / 256;
    bp_trip_direct_kernel<<<tb, 256, 0, stream>>>(diff, ind2, ind3, lambd, zeta,
                                                  eta4, out, nt);
  }
}